// GNNActor_66365834658323
// MI455X (gfx1250) — compile-verified
//
#include <hip/hip_runtime.h>
#include <math.h>

#define NN   100000
#define EE   1000000
#define DD   128
#define HIDD 128
#define RRR  64
#define NFLS 100
#define MAXV 20

typedef float v2f __attribute__((ext_vector_type(2)));
typedef float v8f __attribute__((ext_vector_type(8)));

// Native L2 float atomic (no-return form -> STOREcnt tracked)
__device__ __forceinline__ void atomicAddF(float* p, float v) {
    asm volatile("global_atomic_add_f32 %0, %1, off" :: "v"(p), "v"(v) : "memory");
}

// ---------------------------------------------------------------- utilities
__global__ void zero_f32(float* __restrict__ p, int n) {
    int i = blockIdx.x * blockDim.x + threadIdx.x;
    if (i < n) p[i] = 0.0f;
}

__global__ void deg_init(int* __restrict__ degi, int n) {
    int i = blockIdx.x * blockDim.x + threadIdx.x;
    if (i < n) degi[i] = 1;  // self loop
}

__global__ void deg_count(const int* __restrict__ dst, int* __restrict__ degi, int e) {
    int i = blockIdx.x * blockDim.x + threadIdx.x;
    if (i < e) (void)__hip_atomic_fetch_add(&degi[dst[i]], 1, __ATOMIC_RELAXED, __HIP_MEMORY_SCOPE_AGENT);
}

__global__ void compute_dinv(const int* __restrict__ degi, float* __restrict__ dinv, int n) {
    int i = blockIdx.x * blockDim.x + threadIdx.x;
    if (i < n) dinv[i] = rsqrtf((float)degi[i]);
}

// ---------------------------------------------------------- fp32 WMMA GEMM
// C[M,128] = A[M,K] @ B[K,128] (+bias, optional relu). M%16==0, K%4==0.
// Block = 256 threads (8 waves). B staged once into LDS in WMMA-fragment
// order: Bs[(k0/4)*256 + t*32 + lane] = (B[kk][t*16+n], B[kk+1][t*16+n]),
// kk = k0 + 2*(lane>>4), n = lane&15 -> ds_load_2addr_b64-friendly, no bank
// conflicts. One 16-row M tile per wave: 8 accumulators (64 VGPRs), no spills.
__global__ void wmma_gemm128(const float* __restrict__ A, const float* __restrict__ B,
                             const float* __restrict__ bias, float* __restrict__ C,
                             int M, int K, int relu) {
    extern __shared__ float2 Bs[];                 // (K/4)*8*32 float2 = K*512 bytes
    const int k4steps = K >> 2;

    // cooperative stage of B into LDS (fragment-swizzled)
    for (int idx = threadIdx.x; idx < k4steps * 256; idx += blockDim.x) {
        int k4  = idx >> 8;
        int rem = idx & 255;
        int t   = rem >> 5;
        int L   = rem & 31;
        int kk  = (k4 << 2) + ((L >> 4) << 1);
        int n   = (t << 4) + (L & 15);
        Bs[idx] = make_float2(B[(size_t)kk * 128 + n], B[(size_t)(kk + 1) * 128 + n]);
    }
    __syncthreads();

    int lane   = threadIdx.x & 31;
    int mtiles = M >> 4;
    int tile   = (int)(blockIdx.x * 8 + (threadIdx.x >> 5));
    if (tile >= mtiles) return;                    // whole-wave exit (EXEC all-1 kept)

    int khalf = (lane >> 4) << 1;                  // lanes 16..31 hold K+2,K+3
    int n     = lane & 15;
    const float* arow = A + (size_t)((tile << 4) + n) * K;

#pragma unroll
    for (int j = 0; j < 4; ++j)                    // warm A row (global_prefetch_b8)
        __builtin_prefetch(arow + j * 32, 0, 0);

    v8f acc[8] = {};
    for (int k0 = 0; k0 < K; k0 += 4) {
        int kk = k0 + khalf;                       // even -> 8B-aligned pair
        float2 av = *(const float2*)(arow + kk);
        v2f a; a.x = av.x; a.y = av.y;
        const float2* bp = Bs + ((size_t)(k0 >> 2) << 8) + lane;
#pragma unroll
        for (int t = 0; t < 8; ++t) {
            float2 bv = bp[t << 5];
            v2f b; b.x = bv.x; b.y = bv.y;
            acc[t] = __builtin_amdgcn_wmma_f32_16x16x4_f32(
                false, a, false, b, (short)0, acc[t], false, false);
        }
    }

    // C/D layout: VGPR i -> row i (lanes 0-15) / row i+8 (lanes 16-31), col = lane&15
    int rbase = (tile << 4) + ((lane >> 4) << 3);
#pragma unroll
    for (int t = 0; t < 8; ++t) {
        int col  = t * 16 + n;
        float bv = bias ? bias[col] : 0.0f;
#pragma unroll
        for (int i = 0; i < 8; ++i) {
            float v = acc[t][i] + bv;
            if (relu) v = fmaxf(v, 0.0f);
            C[(size_t)(rbase + i) * 128 + col] = v;
        }
    }
}

// ------------------------------------------------- GCN edge scatter (atomics)
// One wave per edge; lane moves a float4 (128 floats per edge).
__global__ void edge_scatter(const float* __restrict__ h, const int* __restrict__ src,
                             const int* __restrict__ dst, const float* __restrict__ dinv,
                             float* __restrict__ out, int e) {
    int eid  = (int)((blockIdx.x * blockDim.x + threadIdx.x) >> 5);
    int lane = threadIdx.x & 31;
    if (eid >= e) return;
    int s = src[eid], d = dst[eid];
    float nrm = dinv[s] * dinv[d];
    float4 hv = *(const float4*)(h + (size_t)s * 128 + lane * 4);
    float* o  = out + (size_t)d * 128 + lane * 4;
    atomicAddF(o + 0, hv.x * nrm);
    atomicAddF(o + 1, hv.y * nrm);
    atomicAddF(o + 2, hv.z * nrm);
    atomicAddF(o + 3, hv.w * nrm);
}

// self-loop term + bias (+relu), in place on agg
__global__ void gcn_finalize(float* __restrict__ agg, const float* __restrict__ hw,
                             const float* __restrict__ dinv, const float* __restrict__ bias,
                             int relu, int total) {
    int idx = blockIdx.x * blockDim.x + threadIdx.x;
    if (idx >= total) return;
    int i = idx >> 7, c = idx & 127;
    float di = dinv[i];
    float v = agg[idx] + hw[idx] * di * di + bias[c];
    if (relu) v = fmaxf(v, 0.0f);
    agg[idx] = v;
}

// ----------------------------------------------------- segment-mean pooling
__global__ void pool_scatter(const float* __restrict__ h2, const int* __restrict__ batch,
                             float* __restrict__ sums, int* __restrict__ cnt, int n) {
    int node = (int)((blockIdx.x * blockDim.x + threadIdx.x) >> 5);
    int lane = threadIdx.x & 31;
    if (node >= n) return;
    int b = batch[node];
    if (b <= 0) return;                 // uniform per wave
    float4 hv = *(const float4*)(h2 + (size_t)node * 128 + lane * 4);
    float* s = sums + (size_t)b * 128 + lane * 4;
    atomicAddF(s + 0, hv.x);
    atomicAddF(s + 1, hv.y);
    atomicAddF(s + 2, hv.z);
    atomicAddF(s + 3, hv.w);
    if (lane == 0)
        (void)__hip_atomic_fetch_add(&cnt[b], 1, __ATOMIC_RELAXED, __HIP_MEMORY_SCOPE_AGENT);
}

__global__ void build_xcat(const float* __restrict__ h2, const float* __restrict__ sums,
                           const int* __restrict__ cnt, const int* __restrict__ cstat,
                           float* __restrict__ xcat) {
    int idx = blockIdx.x * blockDim.x + threadIdx.x;
    if (idx >= RRR * 356) return;
    int r = idx / 356, c = idx % 356;
    float v;
    if (c < 128)      v = h2[r * 128 + c];
    else if (c < 256) v = sums[r * 128 + (c - 128)] / fmaxf((float)cnt[r], 1.0f);
    else              v = (float)cstat[r * NFLS + (c - 256)];
    xcat[idx] = v;
}

// -------------------------------------------- fused heads + pairwise softmax
__global__ void heads_softmax(const float* __restrict__ x2,
                              const float* __restrict__ cW, const float* __restrict__ cb,
                              const float* __restrict__ rW, const float* __restrict__ rb,
                              float* __restrict__ out) {
    __shared__ float row[128];
    int r = blockIdx.y;
    for (int k = threadIdx.x; k < 128; k += blockDim.x) row[k] = x2[r * 128 + k];
    __syncthreads();
    int p = blockIdx.x * blockDim.x + threadIdx.x;   // pair id 0..2099
    if (p >= 2100) return;
    const float* W; const float* Bv; int ld, col; float* o;
    if (p < 100) { W = cW; Bv = cb; ld = 200;  col = 2 * p;
                   o = out + ((size_t)r * 100 + p) * 2; }
    else         { int u = p - 100; W = rW; Bv = rb; ld = 4000; col = 2 * u;
                   o = out + 12800 + ((size_t)r * 2000 + u) * 2; }
    float l0 = Bv[col], l1 = Bv[col + 1];
    for (int k = 0; k < 128; ++k) {
        float xv = row[k];
        l0 += xv * W[(size_t)k * ld + col];
        l1 += xv * W[(size_t)k * ld + col + 1];
    }
    float m  = fmaxf(l0, l1);
    float e0 = __expf(l0 - m), e1 = __expf(l1 - m);
    float inv = 1.0f / (e0 + e1);
    o[0] = e0 * inv;
    o[1] = e1 * inv;
}

// ---------------------------------------------------------------------------
extern "C" void kernel_launch(void* const* d_in, const int* in_sizes, int n_in,
                              void* d_out, int out_size, void* d_ws, size_t ws_size,
                              hipStream_t stream) {
    const float* x    = (const float*)d_in[0];
    const int*   ei   = (const int*)  d_in[1];
    const int*   bat  = (const int*)  d_in[2];
    const int*   cst  = (const int*)  d_in[3];
    const float* W1   = (const float*)d_in[4];
    const float* b1   = (const float*)d_in[5];
    const float* W2   = (const float*)d_in[6];
    const float* b2   = (const float*)d_in[7];
    const float* fc1W = (const float*)d_in[8];
    const float* fc1b = (const float*)d_in[9];
    const float* fc2W = (const float*)d_in[10];
    const float* fc2b = (const float*)d_in[11];
    const float* cW   = (const float*)d_in[12];
    const float* cb   = (const float*)d_in[13];
    const float* rW   = (const float*)d_in[14];
    const float* rb   = (const float*)d_in[15];
    const int* src = ei;
    const int* dst = ei + EE;
    float* out = (float*)d_out;

    // workspace layout (floats)
    float* buf0 = (float*)d_ws;                       // N*128 : x@W / h@W2
    float* buf1 = buf0 + (size_t)NN * HIDD;           // N*128 : aggregated h / h2
    int*   degi = (int*)(buf1 + (size_t)NN * HIDD);   // N
    float* dinv = (float*)(degi + NN);                // N
    float* sums = dinv + NN;                          // 64*128
    int*   cnt  = (int*)(sums + RRR * HIDD);          // 64
    float* xcat = (float*)(cnt + RRR);                // 64*356
    float* x1   = xcat + RRR * 356;                   // 64*128
    float* x2   = x1 + RRR * HIDD;                    // 64*128

    const int T  = 256;
    const int NH = NN * HIDD;

    // degrees -> dinv
    deg_init    <<<(NN + T - 1) / T, T, 0, stream>>>(degi, NN);
    deg_count   <<<(EE + T - 1) / T, T, 0, stream>>>(dst, degi, EE);
    compute_dinv<<<(NN + T - 1) / T, T, 0, stream>>>(degi, dinv, NN);

    // GEMM launch: 8 waves/block, 1 M-tile/wave
    int gemmBlocksBig = (NN / 16 + 7) / 8;             // 782
    size_t lds128     = (size_t)HIDD * 512;            // 64 KB
    size_t lds356     = (size_t)(2 * HIDD + NFLS) * 512; // 178 KB (<320 KB WGP LDS)

    // GCN layer 1
    wmma_gemm128<<<gemmBlocksBig, T, lds128, stream>>>(x, W1, nullptr, buf0, NN, DD, 0);
    zero_f32    <<<(NH + T - 1) / T, T, 0, stream>>>(buf1, NH);
    edge_scatter<<<(EE * 32 + T - 1) / T, T, 0, stream>>>(buf0, src, dst, dinv, buf1, EE);
    gcn_finalize<<<(NH + T - 1) / T, T, 0, stream>>>(buf1, buf0, dinv, b1, 1, NH);

    // GCN layer 2
    wmma_gemm128<<<gemmBlocksBig, T, lds128, stream>>>(buf1, W2, nullptr, buf0, NN, HIDD, 0);
    zero_f32    <<<(NH + T - 1) / T, T, 0, stream>>>(buf1, NH);
    edge_scatter<<<(EE * 32 + T - 1) / T, T, 0, stream>>>(buf0, src, dst, dinv, buf1, EE);
    gcn_finalize<<<(NH + T - 1) / T, T, 0, stream>>>(buf1, buf0, dinv, b2, 0, NH);

    // pooling + feature concat (sums and cnt contiguous -> one zero pass)
    zero_f32    <<<(RRR * HIDD + RRR + T - 1) / T, T, 0, stream>>>(sums, RRR * HIDD + RRR);
    pool_scatter<<<(NN * 32 + T - 1) / T, T, 0, stream>>>(buf1, bat, sums, cnt, NN);
    build_xcat  <<<(RRR * 356 + T - 1) / T, T, 0, stream>>>(buf1, sums, cnt, cst, xcat);

    // MLP (M=64 -> 4 M-tiles -> 4 active waves, 1 block each)
    wmma_gemm128<<<1, T, lds356, stream>>>(xcat, fc1W, fc1b, x1, RRR, 2 * HIDD + NFLS, 1);
    wmma_gemm128<<<1, T, lds128, stream>>>(x1,   fc2W, fc2b, x2, RRR, HIDD, 1);

    // heads + softmax -> d_out (12800 cache + 256000 rec)
    dim3 hg((2100 + T - 1) / T, RRR);
    heads_softmax<<<hg, T, 0, stream>>>(x2, cW, cb, rW, rb, out);
}